// GEChebNet_81140522156569
// MI455X (gfx1250) — compile-verified
//
#include <hip/hip_runtime.h>
#include <hip/hip_bf16.h>

typedef __attribute__((ext_vector_type(2))) float v2f;
typedef __attribute__((ext_vector_type(8))) float v8f;

#define VN    20000
#define EN    320000
#define KCH   6
#define BB    8
#define CIN_  3
#define CINP  4              // padded input channels (zero-filled 4th)
#define H0_   16
#define H1_   32
#define H2_   64
#define COUT_ 10
#define COUTP 16             // padded output channels (zero weights/bias)
#define MROWS (VN * BB)      // 160000, divisible by 16
#define BN_EPS 1e-5f

// ---------------------------------------------------------------------------
// (B, 3, V) -> (V, B, 4) flat, channel 3 zero-padded
__global__ void k_transpose_in(const float* __restrict__ x, float* __restrict__ X0) {
    int i = blockIdx.x * blockDim.x + threadIdx.x;
    int total = VN * BB * CINP;
    if (i >= total) return;
    int c = i % CINP;
    int t = i / CINP;
    int b = t % BB;
    int v = t / BB;
    X0[i] = (c < CIN_) ? x[((size_t)b * CIN_ + c) * VN + v] : 0.0f;
}

// pad W_out (K,64,10) -> (K,64,16), b_out (10) -> (16), zeros in pad columns
__global__ void k_pad_wout(const float* __restrict__ W, const float* __restrict__ b,
                           float* __restrict__ Wp, float* __restrict__ bp) {
    int i = blockIdx.x * blockDim.x + threadIdx.x;
    int total = KCH * H2_ * COUTP;
    if (i < total) {
        int o = i % COUTP;
        int t = i / COUTP;            // t = k*H2_ + row
        Wp[i] = (o < COUT_) ? W[(size_t)t * COUT_ + o] : 0.0f;
    }
    if (i < COUTP) bp[i] = (i < COUT_) ? b[i] : 0.0f;
}

// Xn = c0 * Xp  (c0 = 0 -> zero init, c0 = -1 -> start of 2*L*Xc - Xp)
__global__ void k_spmm_init(float* __restrict__ Xn, const float* __restrict__ Xp,
                            float c0, int n) {
    int i = blockIdx.x * blockDim.x + threadIdx.x;
    if (i >= n) return;
    Xn[i] = (c0 == 0.0f) ? 0.0f : c0 * Xp[i];
}

// Xn[rows[e]*F + f] += s * vals[e] * Xc[cols[e]*F + f]
__global__ void k_spmm_edges(const int* __restrict__ rows, const int* __restrict__ cols,
                             const float* __restrict__ vals, const float* __restrict__ Xc,
                             float* __restrict__ Xn, int F, int chunks, float s) {
    int e = blockIdx.x / chunks;
    int f = (blockIdx.x % chunks) * blockDim.x + threadIdx.x;
    if (f >= F) return;
    int r = rows[e];
    int c = cols[e];
    float a = s * vals[e] * Xc[(size_t)c * F + f];
    unsafeAtomicAdd(&Xn[(size_t)r * F + f], a);
}

// ---------------------------------------------------------------------------
// acc(+)= X (MROWS x CIN) * Wk (CIN x COUT) via V_WMMA_F32_16X16X4_F32.
// Compile-time shapes, CIN % 4 == 0, COUT % 16 == 0: no predication anywhere.
// One wave per 16x16 output tile.
// MODE: 0 = overwrite acc, 1 = accumulate into acc, 2 = accumulate+bias+relu
template <int CIN, int COUT, int MODE>
__global__ void __launch_bounds__(256)
k_proj_wmma(const float* __restrict__ X, const float* __restrict__ Wk,
            float* __restrict__ acc, const float* __restrict__ bias) {
    const int lane = threadIdx.x & 31;
    const int wave = threadIdx.x >> 5;
    const int mtiles = MROWS / 16;
    const int tile = blockIdx.x * 8 + wave;      // grid sized exactly: no guard
    const int mt = tile % mtiles;
    const int nt = tile / mtiles;
    const int row0 = mt * 16;
    const int l16  = lane & 15;
    const int hi   = lane >> 4;                  // selects K pair {0,1} vs {2,3}
    const int nn   = nt * 16 + l16;              // this lane's output column

    // per-lane base pointers; all further offsets are compile-time immediates
    const float* bp = Wk + (size_t)(2 * hi) * COUT + nn;        // B operand
    const float* ap = X + (size_t)(row0 + l16) * CIN + 2 * hi;  // A operand
    float*       cp = acc + (size_t)(row0 + 8 * hi) * COUT + nn;

    v8f c = {};
    if (MODE != 0) {
#pragma unroll
        for (int j = 0; j < 8; ++j) c[j] = cp[(size_t)j * COUT];
    }

#pragma unroll
    for (int kb = 0; kb < CIN; kb += 4) {
        v2f a = *(const v2f*)(ap + kb);          // K = kb+2*hi, kb+2*hi+1 (b64)
        v2f b;
        b.x = bp[(size_t)kb * COUT];             // row kb+2*hi
        b.y = bp[(size_t)(kb + 1) * COUT];       // row kb+2*hi+1
        c = __builtin_amdgcn_wmma_f32_16x16x4_f32(
                false, a, false, b, (short)0, c, false, false);
    }

    if (MODE == 2) {
        float bv = bias[nn];
#pragma unroll
        for (int j = 0; j < 8; ++j)
            cp[(size_t)j * COUT] = fmaxf(c[j] + bv, 0.0f);
    } else {
#pragma unroll
        for (int j = 0; j < 8; ++j)
            cp[(size_t)j * COUT] = c[j];
    }
}

// ---------------------------------------------------------------------------
// per-channel mean / inv-std over MROWS samples (training-mode BN, biased var)
__global__ void k_bn_stats(const float* __restrict__ Z, float* __restrict__ mean,
                           float* __restrict__ invstd, int C) {
    int c = blockIdx.x;
    float s = 0.0f, sq = 0.0f;
    for (int r = threadIdx.x; r < MROWS; r += blockDim.x) {
        float v = Z[(size_t)r * C + c];
        s += v;
        sq += v * v;
    }
    __shared__ float sh_s[256];
    __shared__ float sh_q[256];
    sh_s[threadIdx.x] = s;
    sh_q[threadIdx.x] = sq;
    __syncthreads();
    for (int off = blockDim.x / 2; off > 0; off >>= 1) {
        if ((int)threadIdx.x < off) {
            sh_s[threadIdx.x] += sh_s[threadIdx.x + off];
            sh_q[threadIdx.x] += sh_q[threadIdx.x + off];
        }
        __syncthreads();
    }
    if (threadIdx.x == 0) {
        float m   = sh_s[0] / (float)MROWS;
        float var = sh_q[0] / (float)MROWS - m * m;
        mean[c]   = m;
        invstd[c] = rsqrtf(var + BN_EPS);
    }
}

__global__ void k_bn_apply(const float* __restrict__ Z, float* __restrict__ Y,
                           const float* __restrict__ g, const float* __restrict__ be,
                           const float* __restrict__ mean, const float* __restrict__ invstd,
                           int C, int n) {
    int i = blockIdx.x * blockDim.x + threadIdx.x;
    if (i >= n) return;
    int c = i % C;
    Y[i] = g[c] * (Z[i] - mean[c]) * invstd[c] + be[c];
}

// ---------------------------------------------------------------------------
// final acc layout: (V, B, COUTP=16); max over V for o < 10
__global__ void k_pool_max(const float* __restrict__ Z, float* __restrict__ pooled) {
    int b = blockIdx.x / COUT_;
    int o = blockIdx.x % COUT_;
    float m = -3.402823466e38f;
    for (int v = threadIdx.x; v < VN; v += blockDim.x)
        m = fmaxf(m, Z[((size_t)v * BB + b) * COUTP + o]);
    __shared__ float sh[256];
    sh[threadIdx.x] = m;
    __syncthreads();
    for (int off = blockDim.x / 2; off > 0; off >>= 1) {
        if ((int)threadIdx.x < off)
            sh[threadIdx.x] = fmaxf(sh[threadIdx.x], sh[threadIdx.x + off]);
        __syncthreads();
    }
    if (threadIdx.x == 0) pooled[blockIdx.x] = sh[0];
}

__global__ void k_logsoftmax(const float* __restrict__ pooled, float* __restrict__ out) {
    int b = threadIdx.x;
    if (b >= BB) return;
    float m = -3.402823466e38f;
    for (int o = 0; o < COUT_; ++o) m = fmaxf(m, pooled[b * COUT_ + o]);
    float s = 0.0f;
    for (int o = 0; o < COUT_; ++o) s += expf(pooled[b * COUT_ + o] - m);
    float lse = m + logf(s);
    for (int o = 0; o < COUT_; ++o) out[b * COUT_ + o] = pooled[b * COUT_ + o] - lse;
}

// ---------------------------------------------------------------------------
struct Ctx {
    const int* rows;
    const int* cols;
    const float* vals;
    float* bufA;
    float* bufB;
    float* bufC;
    float* acc;
    hipStream_t stream;
};

template <int CINT, int COUTT>
static void run_layer_t(const Ctx& c, const float* Wl, const float* bias) {
    const int F = BB * CINT;
    float* Xp = c.bufA;
    float* Xc = c.bufB;
    float* Xn = c.bufC;
    constexpr int blocks = (MROWS / 16) * (COUTT / 16) / 8;   // exact, no tail

    auto proj = [&](const float* Xm, int k, int mode) {
        const float* Wkp = Wl + (size_t)k * CINT * COUTT;
        if (mode == 0)
            k_proj_wmma<CINT, COUTT, 0><<<blocks, 256, 0, c.stream>>>(
                Xm, Wkp, c.acc, bias);
        else if (mode == 1)
            k_proj_wmma<CINT, COUTT, 1><<<blocks, 256, 0, c.stream>>>(
                Xm, Wkp, c.acc, bias);
        else
            k_proj_wmma<CINT, COUTT, 2><<<blocks, 256, 0, c.stream>>>(
                Xm, Wkp, c.acc, bias);
    };
    auto spmm = [&](const float* Xcur, const float* Xprev, float* Xnew,
                    float c0, float s) {
        int n = VN * F;
        k_spmm_init<<<(n + 255) / 256, 256, 0, c.stream>>>(
            Xnew, Xprev ? Xprev : Xnew, c0, n);
        int tb = (F < 256) ? F : 256;             // F in {32,128,256,512}
        int chunks = (F + tb - 1) / tb;
        k_spmm_edges<<<EN * chunks, tb, 0, c.stream>>>(
            c.rows, c.cols, c.vals, Xcur, Xnew, F, chunks, s);
    };

    proj(Xp, 0, /*mode=*/0);                      // k=0 term (X0)
    spmm(Xp, nullptr, Xc, 0.0f, 1.0f);            // X1 = L X0
    proj(Xc, 1, 1);
    for (int k = 2; k < KCH; ++k) {
        spmm(Xc, Xp, Xn, -1.0f, 2.0f);            // Xn = 2 L Xc - Xp
        proj(Xn, k, (k == KCH - 1) ? 2 : 1);
        float* t = Xp; Xp = Xc; Xc = Xn; Xn = t;
    }
}

extern "C" void kernel_launch(void* const* d_in, const int* in_sizes, int n_in,
                              void* d_out, int out_size, void* d_ws, size_t ws_size,
                              hipStream_t stream) {
    const float* x         = (const float*)d_in[0];
    const float* edge_vals = (const float*)d_in[1];
    const float* W_in      = (const float*)d_in[2];
    const float* b_in      = (const float*)d_in[3];
    const float* g0        = (const float*)d_in[4];
    const float* be0       = (const float*)d_in[5];
    const float* W0        = (const float*)d_in[6];
    const float* bw0       = (const float*)d_in[7];
    const float* g1        = (const float*)d_in[8];
    const float* be1       = (const float*)d_in[9];
    const float* W1        = (const float*)d_in[10];
    const float* bw1       = (const float*)d_in[11];
    const float* g_out     = (const float*)d_in[12];
    const float* be_out    = (const float*)d_in[13];
    const float* W_out     = (const float*)d_in[14];
    const float* b_out     = (const float*)d_in[15];
    const int*   rows      = (const int*)d_in[16];
    const int*   cols      = (const int*)d_in[17];
    float*       out       = (float*)d_out;

    const size_t BUF = (size_t)VN * 512;          // floats per big buffer
    float* bufA = (float*)d_ws;
    float* bufB = bufA + BUF;
    float* bufC = bufB + BUF;
    float* acc  = bufC + BUF;
    float* meanp   = acc + BUF;
    float* invstdp = meanp + 64;
    float* pooled  = invstdp + 64;
    float* Wpad    = pooled + 128;                // KCH*H2_*COUTP = 6144 floats
    float* bpad    = Wpad + KCH * H2_ * COUTP;    // 16 floats

    Ctx ctx{rows, cols, edge_vals, bufA, bufB, bufC, acc, stream};

    auto bn = [&](const float* g, const float* be, int C) {
        k_bn_stats<<<C, 256, 0, stream>>>(acc, meanp, invstdp, C);
        int n = MROWS * C;
        k_bn_apply<<<(n + 255) / 256, 256, 0, stream>>>(acc, bufA, g, be,
                                                        meanp, invstdp, C, n);
    };

    k_transpose_in<<<(VN * BB * CINP + 255) / 256, 256, 0, stream>>>(x, bufA);
    k_pad_wout<<<(KCH * H2_ * COUTP + 255) / 256, 256, 0, stream>>>(
        W_out, b_out, Wpad, bpad);

    run_layer_t<CINP, H0_>(ctx, W_in, b_in);      // acc = relu(cheb0), (V,B,16)
    bn(g0, be0, H0_);                             // bufA = BN output
    run_layer_t<H0_, H1_>(ctx, W0, bw0);
    bn(g1, be1, H1_);
    run_layer_t<H1_, H2_>(ctx, W1, bw1);
    bn(g_out, be_out, H2_);
    run_layer_t<H2_, COUTP>(ctx, Wpad, bpad);     // acc = relu(cheb3), (V,B,16)

    k_pool_max<<<BB * COUT_, 256, 0, stream>>>(acc, pooled);
    k_logsoftmax<<<1, 32, 0, stream>>>(pooled, out);
}